// TemporalMemoryBank_59889023976174
// MI455X (gfx1250) — compile-verified
//
#include <hip/hip_runtime.h>
#include <math.h>

// Problem constants (match reference).
#define B_SZ 512
#define M_SZ 32768
#define D_SZ 256
#define H_SZ 4
#define HD_SZ 64

typedef __attribute__((ext_vector_type(2))) float v2f;
typedef __attribute__((ext_vector_type(8))) float v8f;

// D = A(16x4, f32) * B(4x16, f32) + C(16x16, f32) on the wave32 matrix pipe.
__device__ __forceinline__ v8f wmma_f32_4(v2f a, v2f b, v8f c) {
  return __builtin_amdgcn_wmma_f32_16x16x4_f32(
      /*neg_a=*/false, a, /*neg_b=*/false, b,
      /*c_mod=*/(short)0, c, /*reuse_a=*/false, /*reuse_b=*/false);
}

// ---------------------------------------------------------------------------
// C[N,256] = (A[N,256] @ W[256,256] + bias) * scale
// One 16x16 output tile per wave, 8 waves per block.
// A fragment (16x4): lane l -> row = l&15, k = 2*(l>>4)+{0,1} (float2 load).
// B fragment (4x16): lane l -> col = l&15, k = 2*(l>>4)+{0,1}.
// C tile: VGPR v, lanes 0-15 -> (row v, col lane); lanes 16-31 -> (row v+8, ...).
// ---------------------------------------------------------------------------
__global__ void gemm_bias_kernel(const float* __restrict__ A,
                                 const float* __restrict__ W,
                                 const float* __restrict__ bias,
                                 float* __restrict__ C, int N, float scale) {
  const int lane = threadIdx.x & 31;
  const int wave = threadIdx.x >> 5;
  const int c = lane & 15;
  const int kh = (lane >> 4) << 1;
  const int tile = blockIdx.x * 8 + wave;
  const int r0 = (tile >> 4) << 4;
  const int c0 = (tile & 15) << 4;
  if (r0 >= N) return;

  v8f acc = {};
  const float* arow = A + (size_t)(r0 + c) * D_SZ;
  for (int k = 0; k < D_SZ; k += 4) {
    v2f a = *(const v2f*)(arow + k + kh);
    v2f b;
    b.x = W[(size_t)(k + kh) * D_SZ + c0 + c];
    b.y = W[(size_t)(k + kh + 1) * D_SZ + c0 + c];
    acc = wmma_f32_4(a, b, acc);
  }
  const float bb = bias[c0 + c];
  const int rbase = r0 + ((lane >> 4) << 3);
#pragma unroll
  for (int v = 0; v < 8; ++v) {
    C[(size_t)(rbase + v) * D_SZ + c0 + c] = (acc[v] + bb) * scale;
  }
}

// ---------------------------------------------------------------------------
// Pass 1: per (b,h) softmax statistics (running max, sum of exp) over M.
// S^T tile: A = 16 K-rows x 4k, B = Q^T 4k x 16q => per-lane s values run
// along VGPRs (m dim), so the reduction is per-lane scalar + small combine.
// Grid (B/16, H), 8 waves, each wave covers M/8 rows.
// ---------------------------------------------------------------------------
__global__ void attn_stats_kernel(const float* __restrict__ Qs,
                                  const float* __restrict__ K,
                                  float* __restrict__ mxo,
                                  float* __restrict__ zo) {
  const int lane = threadIdx.x & 31;
  const int wave = threadIdx.x >> 5;
  const int c = lane & 15;
  const int kh = (lane >> 4) << 1;
  const int q0 = blockIdx.x << 4;
  const int h = blockIdx.y;
  const int hb = h * HD_SZ;

  v2f qb[16];
  const float* qrow = Qs + (size_t)(q0 + c) * D_SZ + hb;
#pragma unroll
  for (int t = 0; t < 16; ++t) qb[t] = *(const v2f*)(qrow + 4 * t + kh);

  float rmax = -INFINITY, rz = 0.f;
  const int mstart = wave * (M_SZ / 8);
  const int mend = mstart + (M_SZ / 8);
  for (int m0 = mstart; m0 < mend; m0 += 16) {
    v8f s = {};
    const float* krow = K + (size_t)(m0 + c) * D_SZ + hb;
#pragma unroll
    for (int t = 0; t < 16; ++t) {
      v2f a = *(const v2f*)(krow + 4 * t + kh);
      s = wmma_f32_4(a, qb[t], s);
    }
    float cm = s[0];
#pragma unroll
    for (int v = 1; v < 8; ++v) cm = fmaxf(cm, s[v]);
    if (cm > rmax) { rz *= __expf(rmax - cm); rmax = cm; }
#pragma unroll
    for (int v = 0; v < 8; ++v) rz += __expf(s[v] - rmax);
  }

  __shared__ float smx[256];
  __shared__ float sz[256];
  smx[threadIdx.x] = rmax;
  sz[threadIdx.x] = rz;
  __syncthreads();
  if (threadIdx.x < 16) {
    const int q = threadIdx.x;
    float gm = -INFINITY;
    for (int p = 0; p < 16; ++p) {
      int idx = (p >> 1) * 32 + (p & 1) * 16 + q;
      gm = fmaxf(gm, smx[idx]);
    }
    float Z = 0.f;
    for (int p = 0; p < 16; ++p) {
      int idx = (p >> 1) * 32 + (p & 1) * 16 + q;
      Z += sz[idx] * __expf(smx[idx] - gm);
    }
    mxo[(q0 + q) * H_SZ + h] = gm;
    zo[(q0 + q) * H_SZ + h] = Z;
  }
}

// ---------------------------------------------------------------------------
// Pass 2: O[b,h,:] = sum_m softmax(s)[m] * V[m,h,:].
// Recomputes S^T, normalizes with the final (mx, Z), stages the 16x16 P tile
// through per-wave LDS to re-layout into a WMMA-A fragment for P·V.
// ---------------------------------------------------------------------------
__global__ void attn_out_kernel(const float* __restrict__ Qs,
                                const float* __restrict__ K,
                                const float* __restrict__ V,
                                const float* __restrict__ mx,
                                const float* __restrict__ zz,
                                float* __restrict__ Opre) {
  const int lane = threadIdx.x & 31;
  const int wave = threadIdx.x >> 5;
  const int c = lane & 15;
  const int hi = lane >> 4;
  const int kh = hi << 1;
  const int q0 = blockIdx.x << 4;
  const int h = blockIdx.y;
  const int hb = h * HD_SZ;

  v2f qb[16];
  const float* qrow = Qs + (size_t)(q0 + c) * D_SZ + hb;
#pragma unroll
  for (int t = 0; t < 16; ++t) qb[t] = *(const v2f*)(qrow + 4 * t + kh);

  const float mq = mx[(q0 + c) * H_SZ + h];
  const float iz = 1.f / zz[(q0 + c) * H_SZ + h];

  v8f o[4] = {};
  __shared__ float plds[8][256];  // per-wave 16x16 P staging
  float* pw = plds[wave];

  const int mstart = wave * (M_SZ / 8);
  const int mend = mstart + (M_SZ / 8);
  for (int m0 = mstart; m0 < mend; m0 += 16) {
    v8f s = {};
    const float* krow = K + (size_t)(m0 + c) * D_SZ + hb;
#pragma unroll
    for (int t = 0; t < 16; ++t) {
      v2f a = *(const v2f*)(krow + 4 * t + kh);
      s = wmma_f32_4(a, qb[t], s);
    }
    // normalized attention weights: lane holds P[m = m0+v+8*hi][q = c]
#pragma unroll
    for (int v = 0; v < 8; ++v)
      pw[(v + hi * 8) * 16 + c] = __expf(s[v] - mq) * iz;
    __syncthreads();
    // O(16q x 64d) += P(16q x 16m) * V(16m x 64d), 4 m-subchunks x 4 d-tiles
#pragma unroll
    for (int sub = 0; sub < 4; ++sub) {
      const int ms = sub * 4;
      v2f a;
      a.x = pw[(ms + kh) * 16 + c];
      a.y = pw[(ms + kh + 1) * 16 + c];
      const float* vrow = V + (size_t)(m0 + ms + kh) * D_SZ + hb;
#pragma unroll
      for (int dt = 0; dt < 4; ++dt) {
        v2f b;
        b.x = vrow[dt * 16 + c];
        b.y = vrow[D_SZ + dt * 16 + c];
        o[dt] = wmma_f32_4(a, b, o[dt]);
      }
    }
    __syncthreads();
  }

  // combine partial O across the 8 waves
  __shared__ float os[16 * 64];
  for (int i = threadIdx.x; i < 1024; i += 256) os[i] = 0.f;
  __syncthreads();
#pragma unroll
  for (int dt = 0; dt < 4; ++dt)
#pragma unroll
    for (int v = 0; v < 8; ++v)
      atomicAdd(&os[(v + hi * 8) * 64 + dt * 16 + c], o[dt][v]);
  __syncthreads();
  for (int i = threadIdx.x; i < 1024; i += 256) {
    const int q = i >> 6, d = i & 63;
    Opre[(size_t)(q0 + q) * D_SZ + hb + d] = os[i];
  }
}

// ---------------------------------------------------------------------------
// Pass 3: argmax_m of mean_h attn[b,h,m], via recomputed scores.
// Grid (B/16, M/512); packed (f32 bits << 32 | ~m) u64 atomicMax so ties
// pick the smallest m (matches jnp.argmax first-hit semantics).
// ---------------------------------------------------------------------------
__global__ void argmax_kernel(const float* __restrict__ Qs,
                              const float* __restrict__ K,
                              const float* __restrict__ mx,
                              const float* __restrict__ zz,
                              unsigned long long* __restrict__ argbuf) {
  const int lane = threadIdx.x & 31;
  const int wave = threadIdx.x >> 5;
  const int c = lane & 15;
  const int hi = lane >> 4;
  const int kh = hi << 1;
  const int q0 = blockIdx.x << 4;
  const int mbase = blockIdx.y * 512 + wave * 64;

  float bestv = -1.f;
  int bestm = 0;
  for (int ch = 0; ch < 4; ++ch) {
    const int m0 = mbase + ch * 16;
    float w8[8];
#pragma unroll
    for (int v = 0; v < 8; ++v) w8[v] = 0.f;
    for (int h = 0; h < H_SZ; ++h) {
      const int hb = h * HD_SZ;
      const float mq = mx[(q0 + c) * H_SZ + h];
      const float iz = 0.25f / zz[(q0 + c) * H_SZ + h];
      const float* qrow = Qs + (size_t)(q0 + c) * D_SZ + hb;
      const float* krow = K + (size_t)(m0 + c) * D_SZ + hb;
      v8f s = {};
#pragma unroll
      for (int t = 0; t < 16; ++t) {
        v2f a = *(const v2f*)(krow + 4 * t + kh);
        v2f b = *(const v2f*)(qrow + 4 * t + kh);
        s = wmma_f32_4(a, b, s);
      }
#pragma unroll
      for (int v = 0; v < 8; ++v) w8[v] += __expf(s[v] - mq) * iz;
    }
#pragma unroll
    for (int v = 0; v < 8; ++v) {
      if (w8[v] > bestv) { bestv = w8[v]; bestm = m0 + v + hi * 8; }
    }
  }
  // merge the two lane-halves holding the same q
  float ov = __shfl_xor(bestv, 16, 32);
  int om = __shfl_xor(bestm, 16, 32);
  if (ov > bestv || (ov == bestv && om < bestm)) { bestv = ov; bestm = om; }

  __shared__ float svals[8][16];
  __shared__ int sms[8][16];
  if (lane < 16) { svals[wave][lane] = bestv; sms[wave][lane] = bestm; }
  __syncthreads();
  if (threadIdx.x < 16) {
    const int q = threadIdx.x;
    float bv = svals[0][q];
    int bm = sms[0][q];
    for (int w = 1; w < 8; ++w) {
      const float vv = svals[w][q];
      const int mm = sms[w][q];
      if (vv > bv || (vv == bv && mm < bm)) { bv = vv; bm = mm; }
    }
    const unsigned long long packed =
        ((unsigned long long)__float_as_uint(bv) << 32) |
        (unsigned long long)(0xFFFFFFFFu - (unsigned)bm);
    atomicMax(&argbuf[q0 + q], packed);
  }
}

__global__ void init_argbuf_kernel(unsigned long long* __restrict__ a) {
  const int i = blockIdx.x * blockDim.x + threadIdx.x;
  if (i < B_SZ) a[i] = 0ull;
}

// ---------------------------------------------------------------------------
// Sequential gated scatter-update: one workgroup preserves the b-ordering
// (later batch items must see earlier writes to the same memory row).
// ---------------------------------------------------------------------------
__global__ void seq_update_kernel(const float* __restrict__ query,
                                  const unsigned long long* __restrict__ argbuf,
                                  const float* __restrict__ Wg,
                                  const float* __restrict__ bg,
                                  float* __restrict__ nmem) {
  __shared__ float oldr[D_SZ];
  __shared__ float qr[D_SZ];
  const int j = threadIdx.x;  // 256 threads, one per output column
  const float bgj = bg[j];
  for (int b = 0; b < B_SZ; ++b) {
    const int idx = (int)(0xFFFFFFFFu - (unsigned)(argbuf[b] & 0xFFFFFFFFull));
    oldr[j] = nmem[(size_t)idx * D_SZ + j];
    qr[j] = query[(size_t)b * D_SZ + j];
    __syncthreads();
    float acc = bgj;
    for (int i = 0; i < D_SZ; ++i) acc = fmaf(oldr[i], Wg[(size_t)i * D_SZ + j], acc);
    for (int i = 0; i < D_SZ; ++i) acc = fmaf(qr[i], Wg[(size_t)(D_SZ + i) * D_SZ + j], acc);
    const float g = 1.f / (1.f + __expf(-acc));
    const float nv = g * qr[j] + (1.f - g) * oldr[j];
    __syncthreads();
    nmem[(size_t)idx * D_SZ + j] = nv;
    __threadfence();
    __syncthreads();
  }
}

extern "C" void kernel_launch(void* const* d_in, const int* in_sizes, int n_in,
                              void* d_out, int out_size, void* d_ws, size_t ws_size,
                              hipStream_t stream) {
  const float* query = (const float*)d_in[0];
  const float* memory = (const float*)d_in[1];
  const float* Wq = (const float*)d_in[2];
  const float* bq = (const float*)d_in[3];
  const float* Wk = (const float*)d_in[4];
  const float* bk = (const float*)d_in[5];
  const float* Wv = (const float*)d_in[6];
  const float* bv = (const float*)d_in[7];
  const float* Wo = (const float*)d_in[8];
  const float* bo = (const float*)d_in[9];
  const float* Wg = (const float*)d_in[10];
  const float* bg = (const float*)d_in[11];

  float* out_retrieved = (float*)d_out;
  float* out_newmem = out_retrieved + (size_t)B_SZ * D_SZ;

  float* ws = (float*)d_ws;
  float* Qs = ws;    ws += (size_t)B_SZ * D_SZ;   // Q projection, pre-scaled 1/8
  float* Kp = ws;    ws += (size_t)M_SZ * D_SZ;
  float* Vp = ws;    ws += (size_t)M_SZ * D_SZ;
  float* Opre = ws;  ws += (size_t)B_SZ * D_SZ;   // attn @ V, pre-Wo
  float* mxs = ws;   ws += (size_t)B_SZ * H_SZ;
  float* zs = ws;    ws += (size_t)B_SZ * H_SZ;
  unsigned long long* argbuf = (unsigned long long*)ws;

  // 1) projections (Q pre-scaled by 1/sqrt(HD) = 1/8)
  gemm_bias_kernel<<<dim3((B_SZ / 16 * 16) / 8), dim3(256), 0, stream>>>(
      query, Wq, bq, Qs, B_SZ, 0.125f);
  gemm_bias_kernel<<<dim3((M_SZ / 16 * 16) / 8), dim3(256), 0, stream>>>(
      memory, Wk, bk, Kp, M_SZ, 1.f);
  gemm_bias_kernel<<<dim3((M_SZ / 16 * 16) / 8), dim3(256), 0, stream>>>(
      memory, Wv, bv, Vp, M_SZ, 1.f);

  // 2) softmax statistics per (b,h)
  attn_stats_kernel<<<dim3(B_SZ / 16, H_SZ), dim3(256), 0, stream>>>(Qs, Kp, mxs, zs);

  // 3) O = attn @ V (per head)
  attn_out_kernel<<<dim3(B_SZ / 16, H_SZ), dim3(256), 0, stream>>>(Qs, Kp, Vp, mxs, zs, Opre);

  // 4) argmax_m of head-averaged attention
  init_argbuf_kernel<<<dim3(2), dim3(256), 0, stream>>>(argbuf);
  argmax_kernel<<<dim3(B_SZ / 16, M_SZ / 512), dim3(256), 0, stream>>>(Qs, Kp, mxs, zs, argbuf);

  // 5) retrieved = O @ Wo + bo
  gemm_bias_kernel<<<dim3((B_SZ / 16 * 16) / 8), dim3(256), 0, stream>>>(
      Opre, Wo, bo, out_retrieved, B_SZ, 1.f);

  // 6) new_memory = copy of memory, then sequential gated update
  hipMemcpyAsync(out_newmem, memory, (size_t)M_SZ * D_SZ * sizeof(float),
                 hipMemcpyDeviceToDevice, stream);
  seq_update_kernel<<<dim3(1), dim3(256), 0, stream>>>(query, argbuf, Wg, bg, out_newmem);
}